// SVAE_LDS_75222057222386
// MI455X (gfx1250) — compile-verified
//
#include <hip/hip_runtime.h>
#include <hip/hip_bf16.h>

#define T_STEPS 2048
#define TP1     2049
#define NB      64
#define HID     20
#define POS     3
#define LOG_2PI 1.8378770664093453f

typedef __attribute__((ext_vector_type(2))) float v2f;
typedef __attribute__((ext_vector_type(8))) float v8f;

// ---------------- wave32 butterfly reductions (no barriers, one wave per chain)
__device__ __forceinline__ float wave_sum(float v) {
#pragma unroll
  for (int o = 16; o > 0; o >>= 1) v += __shfl_xor(v, o, 32);
  return v;
}
__device__ __forceinline__ float wave_max(float v) {
#pragma unroll
  for (int o = 16; o > 0; o >>= 1) v = fmaxf(v, __shfl_xor(v, o, 32));
  return v;
}

// ---------------- 4x4 SPD Cholesky (lower) + solve, fully unrolled in VGPRs
__device__ __forceinline__ void chol4(const float S[4][4], float L[4][4]) {
  L[0][0] = sqrtf(S[0][0]);
  L[1][0] = S[1][0] / L[0][0];
  L[2][0] = S[2][0] / L[0][0];
  L[3][0] = S[3][0] / L[0][0];
  L[1][1] = sqrtf(S[1][1] - L[1][0]*L[1][0]);
  L[2][1] = (S[2][1] - L[2][0]*L[1][0]) / L[1][1];
  L[3][1] = (S[3][1] - L[3][0]*L[1][0]) / L[1][1];
  L[2][2] = sqrtf(S[2][2] - L[2][0]*L[2][0] - L[2][1]*L[2][1]);
  L[3][2] = (S[3][2] - L[3][0]*L[2][0] - L[3][1]*L[2][1]) / L[2][2];
  L[3][3] = sqrtf(S[3][3] - L[3][0]*L[3][0] - L[3][1]*L[3][1] - L[3][2]*L[3][2]);
  L[0][1] = L[0][2] = L[0][3] = L[1][2] = L[1][3] = L[2][3] = 0.f;
}
__device__ __forceinline__ void chol_solve4(const float L[4][4], const float b[4], float x[4]) {
  float y0 = b[0] / L[0][0];
  float y1 = (b[1] - L[1][0]*y0) / L[1][1];
  float y2 = (b[2] - L[2][0]*y0 - L[2][1]*y1) / L[2][2];
  float y3 = (b[3] - L[3][0]*y0 - L[3][1]*y1 - L[3][2]*y2) / L[3][3];
  x[3] = y3 / L[3][3];
  x[2] = (y2 - L[3][2]*x[3]) / L[2][2];
  x[1] = (y1 - L[2][1]*x[2] - L[3][1]*x[3]) / L[1][1];
  x[0] = (y0 - L[1][0]*x[1] - L[2][0]*x[2] - L[3][0]*x[3]) / L[0][0];
}

// Q = L L^T, L = tril(q_param) with exp() on the diagonal
__device__ __forceinline__ void build_Q(const float* __restrict__ qp, float Q[4][4]) {
  const int ri[10] = {0,1,1,2,2,2,3,3,3,3};
  const int ci[10] = {0,0,1,0,1,2,0,1,2,3};
  float L[4][4] = {{0.f}};
#pragma unroll
  for (int k = 0; k < 10; ++k) {
    float v = qp[k];
    if (ri[k] == ci[k]) v = expf(v);
    L[ri[k]][ci[k]] = v;
  }
#pragma unroll
  for (int i = 0; i < 4; ++i)
#pragma unroll
    for (int j = 0; j < 4; ++j)
      Q[i][j] = L[i][0]*L[j][0] + L[i][1]*L[j][1] + L[i][2]*L[j][2] + L[i][3]*L[j][3];
}

// =====================================================================
// Kernel 1: encoder.  One wave = 32 rows.  Layer 1 (3->20) done with
// V_WMMA_F32_16X16X4_F32 (two 16-row tiles x two 16-col N tiles), bias+relu
// applied in C layout, h staged through wave-local LDS, layer 2 (20->4,4)
// in VALU per lane (one row per lane).
// =====================================================================
__global__ __launch_bounds__(256) void k_encoder(
    const float* __restrict__ x,  const float* __restrict__ w1, const float* __restrict__ b1,
    const float* __restrict__ wx, const float* __restrict__ bx,
    const float* __restrict__ wl, const float* __restrict__ bl,
    float* __restrict__ xhat, float* __restrict__ sigma)
{
  __shared__ float hb[8][32][HID];          // 20.0 KB, wave-private regions
  const int lane = threadIdx.x & 31;
  const int w    = threadIdx.x >> 5;
  const int gw   = blockIdx.x * 8 + w;
  const int nWaves = (TP1 * NB) / 32;       // 4098, exact
  if (gw >= nWaves) return;                 // wave-uniform exit (EXEC stays full)
  const int rbase = gw * 32;

  const int half = lane >> 4;               // K split: lanes 0-15 -> K0/K1, 16-31 -> K2/K3
  const int ml   = lane & 15;
  const int k0 = 2 * half, k1 = 2 * half + 1;

  // A operands: 16x4 f32 tiles (K=3 padded to 4 with zeros)
  v2f a0, a1;
  {
    const int r0 = rbase + ml, r1 = rbase + 16 + ml;
    a0.x = (k0 < POS) ? x[r0 * POS + k0] : 0.f;
    a0.y = (k1 < POS) ? x[r0 * POS + k1] : 0.f;
    a1.x = (k0 < POS) ? x[r1 * POS + k0] : 0.f;
    a1.y = (k1 < POS) ? x[r1 * POS + k1] : 0.f;
  }
  // B operands: 4x16 f32 tiles of W1 (cols 0-15 and 16-31, cols>=20 zero)
  v2f bA, bB;
  {
    const int n0 = ml, n1 = 16 + ml;
    bA.x = (k0 < POS) ? w1[k0 * HID + n0] : 0.f;
    bA.y = (k1 < POS) ? w1[k1 * HID + n0] : 0.f;
    bB.x = (k0 < POS && n1 < HID) ? w1[k0 * HID + n1] : 0.f;
    bB.y = (k1 < POS && n1 < HID) ? w1[k1 * HID + n1] : 0.f;
  }
  v8f c00 = {}, c01 = {}, c10 = {}, c11 = {};
  c00 = __builtin_amdgcn_wmma_f32_16x16x4_f32(false, a0, false, bA, (short)0, c00, false, false);
  c01 = __builtin_amdgcn_wmma_f32_16x16x4_f32(false, a0, false, bB, (short)0, c01, false, false);
  c10 = __builtin_amdgcn_wmma_f32_16x16x4_f32(false, a1, false, bA, (short)0, c10, false, false);
  c11 = __builtin_amdgcn_wmma_f32_16x16x4_f32(false, a1, false, bB, (short)0, c11, false, false);

  // bias + relu in C layout (VGPR i: M = i + 8*half; N = lane&15), stage to LDS
  const float bias0 = b1[ml];
  const float bias1 = (16 + ml < HID) ? b1[16 + ml] : 0.f;
#pragma unroll
  for (int i = 0; i < 8; ++i) {
    const int r = i + 8 * half;
    hb[w][r][ml]      = fmaxf(c00[i] + bias0, 0.f);
    hb[w][16 + r][ml] = fmaxf(c10[i] + bias0, 0.f);
    if (16 + ml < HID) {
      hb[w][r][16 + ml]      = fmaxf(c01[i] + bias1, 0.f);
      hb[w][16 + r][16 + ml] = fmaxf(c11[i] + bias1, 0.f);
    }
  }
  // layer 2: lane -> one row; uniform weight addresses -> scalar loads
  float ax[4] = {0.f, 0.f, 0.f, 0.f}, al[4] = {0.f, 0.f, 0.f, 0.f};
#pragma unroll
  for (int k = 0; k < HID; ++k) {
    const float hv = hb[w][lane][k];
#pragma unroll
    for (int j = 0; j < 4; ++j) {
      ax[j] = fmaf(hv, wx[k * 4 + j], ax[j]);
      al[j] = fmaf(hv, wl[k * 4 + j], al[j]);
    }
  }
  const int row = rbase + lane;
  float4 xo, so;
  xo.x = ax[0] + bx[0]; xo.y = ax[1] + bx[1]; xo.z = ax[2] + bx[2]; xo.w = ax[3] + bx[3];
  so.x = expf(0.5f * (al[0] + bl[0])); so.y = expf(0.5f * (al[1] + bl[1]));
  so.z = expf(0.5f * (al[2] + bl[2])); so.w = expf(0.5f * (al[3] + bl[3]));
  ((float4*)xhat)[row]  = xo;
  ((float4*)sigma)[row] = so;
}

// =====================================================================
// Kernel 2: forward mixture Kalman filter.  One wave32 per chain (64 waves
// total).  Each lane owns 2 of the 64 mixture components; softmax and
// moment matching are wave-butterfly reductions (no LDS, no barriers).
// =====================================================================
__global__ __launch_bounds__(32) void k_filter(
    const float* __restrict__ xhat, const float* __restrict__ sigma,
    const float* __restrict__ Ag, const float* __restrict__ bkf, const float* __restrict__ qp,
    float* __restrict__ fm, float* __restrict__ fP)
{
  const int b = blockIdx.x;
  const int lane = threadIdx.x;
  float A[4][4], bk[4], Q[4][4];
#pragma unroll
  for (int i = 0; i < 4; ++i) {
    bk[i] = bkf[i];
#pragma unroll
    for (int j = 0; j < 4; ++j) A[i][j] = Ag[i * 4 + j];
  }
  build_Q(qp, Q);

  const float4* xh4 = (const float4*)xhat;
  const float4* sg4 = (const float4*)sigma;

  float m[4], P[4][4] = {{0.f}};
  {
    const float4 z0 = xh4[b];
    const float4 s0 = sg4[b];
    m[0] = z0.x; m[1] = z0.y; m[2] = z0.z; m[3] = z0.w;
    P[0][0] = s0.x; P[1][1] = s0.y; P[2][2] = s0.z; P[3][3] = s0.w;
  }
  float* fmb = fm + (size_t)b * T_STEPS * 4;
  float* fPb = fP + (size_t)b * T_STEPS * 16;

  for (int t = 0; t < T_STEPS; ++t) {
    // predict: mp = A m + b ; Pp = A P A^T + Q   (uniform across the wave)
    float mp[4], T1[4][4], Pp[4][4];
#pragma unroll
    for (int i = 0; i < 4; ++i) {
      mp[i] = bk[i] + A[i][0]*m[0] + A[i][1]*m[1] + A[i][2]*m[2] + A[i][3]*m[3];
#pragma unroll
      for (int j = 0; j < 4; ++j)
        T1[i][j] = A[i][0]*P[0][j] + A[i][1]*P[1][j] + A[i][2]*P[2][j] + A[i][3]*P[3][j];
    }
#pragma unroll
    for (int i = 0; i < 4; ++i)
#pragma unroll
      for (int j = 0; j < 4; ++j)
        Pp[i][j] = Q[i][j] + T1[i][0]*A[j][0] + T1[i][1]*A[j][1] + T1[i][2]*A[j][2] + T1[i][3]*A[j][3];

    float ll[2], mi[2][4], Pi[2][4][4];
#pragma unroll
    for (int s = 0; s < 2; ++s) {
      const int n = lane + 32 * s;
      const float4 z  = xh4[(t + 1) * NB + n];
      const float4 rv = sg4[(t + 1) * NB + n];
      float S[4][4];
#pragma unroll
      for (int i = 0; i < 4; ++i)
#pragma unroll
        for (int j = 0; j < 4; ++j) S[i][j] = Pp[i][j];
      S[0][0] += rv.x; S[1][1] += rv.y; S[2][2] += rv.z; S[3][3] += rv.w;
      float L[4][4]; chol4(S, L);
      const float innov[4] = {z.x - mp[0], z.y - mp[1], z.z - mp[2], z.w - mp[3]};
      float sol[4]; chol_solve4(L, innov, sol);
      const float quad = innov[0]*sol[0] + innov[1]*sol[1] + innov[2]*sol[2] + innov[3]*sol[3];
      const float logdet = 2.f * (logf(L[0][0]) + logf(L[1][1]) + logf(L[2][2]) + logf(L[3][3]));
      ll[s] = -0.5f * (quad + logdet + 4.f * LOG_2PI);
      // U = S^{-1} Pp ;  mi = mp + Pp sol ;  Pi = Pp - Pp U  (Pp, S symmetric)
      float U[4][4];
#pragma unroll
      for (int c = 0; c < 4; ++c) {
        const float col[4] = {Pp[0][c], Pp[1][c], Pp[2][c], Pp[3][c]};
        float xc[4]; chol_solve4(L, col, xc);
        U[0][c] = xc[0]; U[1][c] = xc[1]; U[2][c] = xc[2]; U[3][c] = xc[3];
      }
#pragma unroll
      for (int i = 0; i < 4; ++i) {
        mi[s][i] = mp[i] + Pp[i][0]*sol[0] + Pp[i][1]*sol[1] + Pp[i][2]*sol[2] + Pp[i][3]*sol[3];
#pragma unroll
        for (int j = 0; j < 4; ++j)
          Pi[s][i][j] = Pp[i][j] - (Pp[i][0]*U[0][j] + Pp[i][1]*U[1][j] + Pp[i][2]*U[2][j] + Pp[i][3]*U[3][j]);
      }
    }
    // softmax over the 64 components, then moment matching (all butterflies)
    const float Mx = wave_max(fmaxf(ll[0], ll[1]));
    const float e0 = expf(ll[0] - Mx), e1 = expf(ll[1] - Mx);
    const float Zs = wave_sum(e0 + e1);
    const float w0 = e0 / Zs, w1 = e1 / Zs;
    float mu[4];
#pragma unroll
    for (int i = 0; i < 4; ++i) mu[i] = wave_sum(w0 * mi[0][i] + w1 * mi[1][i]);
    float d0[4], d1[4];
#pragma unroll
    for (int i = 0; i < 4; ++i) { d0[i] = mi[0][i] - mu[i]; d1[i] = mi[1][i] - mu[i]; }
    float Pu[4][4];
#pragma unroll
    for (int i = 0; i < 4; ++i)
#pragma unroll
      for (int j = 0; j < 4; ++j)
        Pu[i][j] = wave_sum(w0 * (Pi[0][i][j] + d0[i]*d0[j]) + w1 * (Pi[1][i][j] + d1[i]*d1[j]));
#pragma unroll
    for (int i = 0; i < 4; ++i)
#pragma unroll
      for (int j = i + 1; j < 4; ++j) {
        const float v = 0.5f * (Pu[i][j] + Pu[j][i]);
        Pu[i][j] = v; Pu[j][i] = v;
      }
#pragma unroll
    for (int i = 0; i < 4; ++i) {
      m[i] = mu[i];
#pragma unroll
      for (int j = 0; j < 4; ++j) P[i][j] = Pu[i][j];
    }
    if (lane == 0) {
      float4 mm; mm.x = m[0]; mm.y = m[1]; mm.z = m[2]; mm.w = m[3];
      ((float4*)(fmb + (size_t)t * 4))[0] = mm;
#pragma unroll
      for (int i = 0; i < 4; ++i) {
        float4 pr; pr.x = P[i][0]; pr.y = P[i][1]; pr.z = P[i][2]; pr.w = P[i][3];
        ((float4*)(fPb + (size_t)t * 16 + i * 4))[0] = pr;
      }
    }
    if (t + 2 <= T_STEPS) {   // next step's observations -> global_prefetch
      __builtin_prefetch(&xh4[(t + 2) * NB + lane], 0, 0);
      __builtin_prefetch(&sg4[(t + 2) * NB + lane], 0, 0);
    }
  }
}

// =====================================================================
// Kernel 3: backward RTS smoother fused with sP-Cholesky, reparameterized
// sample and decoder.  One wave32 per chain; sm/sP never hit memory.
// =====================================================================
__device__ __forceinline__ void emit_out(int t, int b,
    const float ms[4], const float Ps[4][4],
    const float* __restrict__ eps, const float* __restrict__ dw,
    const float* __restrict__ db, float* __restrict__ out, int lane)
{
  float Sp[4][4];
#pragma unroll
  for (int i = 0; i < 4; ++i)
#pragma unroll
    for (int j = 0; j < 4; ++j) Sp[i][j] = Ps[i][j];
  Sp[0][0] += 1e-5f; Sp[1][1] += 1e-5f; Sp[2][2] += 1e-5f; Sp[3][3] += 1e-5f;
  float Lc[4][4]; chol4(Sp, Lc);
  const float4 e = ((const float4*)eps)[(size_t)t * NB + b];
  float z[4];
  z[0] = ms[0] + Lc[0][0]*e.x;
  z[1] = ms[1] + Lc[1][0]*e.x + Lc[1][1]*e.y;
  z[2] = ms[2] + Lc[2][0]*e.x + Lc[2][1]*e.y + Lc[2][2]*e.z;
  z[3] = ms[3] + Lc[3][0]*e.x + Lc[3][1]*e.y + Lc[3][2]*e.z + Lc[3][3]*e.w;
  if (lane == 0) {
#pragma unroll
    for (int j = 0; j < 3; ++j) {
      float o = db[j];
#pragma unroll
      for (int i = 0; i < 4; ++i) o = fmaf(z[i], dw[i * 3 + j], o);
      out[((size_t)t * NB + b) * 3 + j] = o;
    }
  }
}

__global__ __launch_bounds__(32) void k_smooth(
    const float* __restrict__ fm, const float* __restrict__ fP,
    const float* __restrict__ eps, const float* __restrict__ dw, const float* __restrict__ db,
    const float* __restrict__ Ag, const float* __restrict__ bkf, const float* __restrict__ qp,
    float* __restrict__ out)
{
  const int b = blockIdx.x;
  const int lane = threadIdx.x;
  float A[4][4], bk[4], Q[4][4];
#pragma unroll
  for (int i = 0; i < 4; ++i) {
    bk[i] = bkf[i];
#pragma unroll
    for (int j = 0; j < 4; ++j) A[i][j] = Ag[i * 4 + j];
  }
  build_Q(qp, Q);
  const float* fmb = fm + (size_t)b * T_STEPS * 4;
  const float* fPb = fP + (size_t)b * T_STEPS * 16;

  float ms[4], Ps[4][4];
  {
    const float4 mm = ((const float4*)(fmb + (size_t)(T_STEPS - 1) * 4))[0];
    ms[0] = mm.x; ms[1] = mm.y; ms[2] = mm.z; ms[3] = mm.w;
#pragma unroll
    for (int i = 0; i < 4; ++i) {
      const float4 pr = ((const float4*)(fPb + (size_t)(T_STEPS - 1) * 16 + i * 4))[0];
      Ps[i][0] = pr.x; Ps[i][1] = pr.y; Ps[i][2] = pr.z; Ps[i][3] = pr.w;
    }
  }
  emit_out(T_STEPS - 1, b, ms, Ps, eps, dw, db, out, lane);

  for (int t = T_STEPS - 2; t >= 0; --t) {
    float mf[4], Pf[4][4];
    {
      const float4 mm = ((const float4*)(fmb + (size_t)t * 4))[0];
      mf[0] = mm.x; mf[1] = mm.y; mf[2] = mm.z; mf[3] = mm.w;
#pragma unroll
      for (int i = 0; i < 4; ++i) {
        const float4 pr = ((const float4*)(fPb + (size_t)t * 16 + i * 4))[0];
        Pf[i][0] = pr.x; Pf[i][1] = pr.y; Pf[i][2] = pr.z; Pf[i][3] = pr.w;
      }
    }
    float mp[4], T1[4][4], Pp[4][4];
#pragma unroll
    for (int i = 0; i < 4; ++i) {
      mp[i] = bk[i] + A[i][0]*mf[0] + A[i][1]*mf[1] + A[i][2]*mf[2] + A[i][3]*mf[3];
#pragma unroll
      for (int j = 0; j < 4; ++j)
        T1[i][j] = A[i][0]*Pf[0][j] + A[i][1]*Pf[1][j] + A[i][2]*Pf[2][j] + A[i][3]*Pf[3][j];
    }
#pragma unroll
    for (int i = 0; i < 4; ++i)
#pragma unroll
      for (int j = 0; j < 4; ++j)
        Pp[i][j] = Q[i][j] + T1[i][0]*A[j][0] + T1[i][1]*A[j][1] + T1[i][2]*A[j][2] + T1[i][3]*A[j][3];

    float L[4][4]; chol4(Pp, L);
    float X[4][4];                       // X = Pp^{-1} (A Pf),  G = X^T
#pragma unroll
    for (int c = 0; c < 4; ++c) {
      const float col[4] = {T1[0][c], T1[1][c], T1[2][c], T1[3][c]};
      float xc[4]; chol_solve4(L, col, xc);
      X[0][c] = xc[0]; X[1][c] = xc[1]; X[2][c] = xc[2]; X[3][c] = xc[3];
    }
    const float dm[4] = {ms[0]-mp[0], ms[1]-mp[1], ms[2]-mp[2], ms[3]-mp[3]};
    float nm[4];
#pragma unroll
    for (int i = 0; i < 4; ++i)          // nm = mf + G dm,  G[i][j] = X[j][i]
      nm[i] = mf[i] + X[0][i]*dm[0] + X[1][i]*dm[1] + X[2][i]*dm[2] + X[3][i]*dm[3];
    float Dp[4][4], GD[4][4], nP[4][4];
#pragma unroll
    for (int i = 0; i < 4; ++i)
#pragma unroll
      for (int j = 0; j < 4; ++j) Dp[i][j] = Ps[i][j] - Pp[i][j];
#pragma unroll
    for (int i = 0; i < 4; ++i)
#pragma unroll
      for (int j = 0; j < 4; ++j)        // GD = G Dp
        GD[i][j] = X[0][i]*Dp[0][j] + X[1][i]*Dp[1][j] + X[2][i]*Dp[2][j] + X[3][i]*Dp[3][j];
#pragma unroll
    for (int i = 0; i < 4; ++i)
#pragma unroll
      for (int j = 0; j < 4; ++j)        // nP = Pf + GD G^T
        nP[i][j] = Pf[i][j] + GD[i][0]*X[0][j] + GD[i][1]*X[1][j] + GD[i][2]*X[2][j] + GD[i][3]*X[3][j];
#pragma unroll
    for (int i = 0; i < 4; ++i)
#pragma unroll
      for (int j = i + 1; j < 4; ++j) {
        const float v = 0.5f * (nP[i][j] + nP[j][i]);
        nP[i][j] = v; nP[j][i] = v;
      }
#pragma unroll
    for (int i = 0; i < 4; ++i) {
      ms[i] = nm[i];
#pragma unroll
      for (int j = 0; j < 4; ++j) Ps[i][j] = nP[i][j];
    }
    emit_out(t, b, ms, Ps, eps, dw, db, out, lane);
    if (t > 0) {
      __builtin_prefetch(fPb + (size_t)(t - 1) * 16, 0, 0);
      __builtin_prefetch(fmb + (size_t)(t - 1) * 4, 0, 0);
    }
  }
}

// =====================================================================
extern "C" void kernel_launch(void* const* d_in, const int* in_sizes, int n_in,
                              void* d_out, int out_size, void* d_ws, size_t ws_size,
                              hipStream_t stream) {
  (void)in_sizes; (void)n_in; (void)out_size; (void)ws_size;
  const float* x   = (const float*)d_in[0];
  const float* eps = (const float*)d_in[1];
  const float* w1  = (const float*)d_in[2];
  const float* b1  = (const float*)d_in[3];
  const float* wx  = (const float*)d_in[4];
  const float* bx  = (const float*)d_in[5];
  const float* wl  = (const float*)d_in[6];
  const float* bl  = (const float*)d_in[7];
  const float* dw  = (const float*)d_in[8];
  const float* db  = (const float*)d_in[9];
  const float* A   = (const float*)d_in[10];
  const float* bkf = (const float*)d_in[11];
  const float* qp  = (const float*)d_in[12];

  float* ws = (float*)d_ws;
  const size_t XH = 0;
  const size_t SG = XH + (size_t)TP1 * NB * 4;
  const size_t FM = SG + (size_t)TP1 * NB * 4;
  const size_t FP = FM + (size_t)NB * T_STEPS * 4;
  float* xhat = ws + XH;
  float* sig  = ws + SG;
  float* fm   = ws + FM;
  float* fP   = ws + FP;

  const int nWaves = (TP1 * NB) / 32;        // 4098 waves, 32 rows each
  const int blocks = (nWaves + 7) / 8;       // 8 waves / block
  k_encoder<<<blocks, 256, 0, stream>>>(x, w1, b1, wx, bx, wl, bl, xhat, sig);
  k_filter <<<NB, 32, 0, stream>>>(xhat, sig, A, bkf, qp, fm, fP);
  k_smooth <<<NB, 32, 0, stream>>>(fm, fP, eps, dw, db, A, bkf, qp, (float*)d_out);
}